// NGCFLayer_73289321939137
// MI455X (gfx1250) — compile-verified
//
#include <hip/hip_runtime.h>

typedef __attribute__((ext_vector_type(2))) float v2f;
typedef __attribute__((ext_vector_type(8))) float v8f;

#define D 64
#define ALPHA 0.2f
#define EPS 1e-12f
#define TILES_PER_BLOCK 8

// ---------------- zero scratch ----------------
__global__ void zero_kernel(float* __restrict__ p, int n) {
    int i = blockIdx.x * blockDim.x + threadIdx.x;
    int stride = gridDim.x * blockDim.x;
    for (; i < n; i += stride) p[i] = 0.0f;
}

// ---------------- SpMM scatter: agg[dst[e]] += w[e] * feat[src[e]] ----------------
// one wave (32 lanes) per edge, float2 per lane (64 feats)
__global__ __launch_bounds__(256)
void spmm_kernel(const float* __restrict__ feat,
                 const int* __restrict__ src,
                 const int* __restrict__ dst,
                 const float* __restrict__ w,
                 float* __restrict__ agg, int nE) {
    int e = blockIdx.x * (blockDim.x >> 5) + (threadIdx.x >> 5);
    int lane = threadIdx.x & 31;
    if (e >= nE) return;
    int s = src[e];
    int d = dst[e];
    float we = w[e];
    float2 v = ((const float2*)(feat + (size_t)s * D))[lane];
    float* a = agg + (size_t)d * D + lane * 2;
    atomicAdd(a,     we * v.x);
    atomicAdd(a + 1, we * v.y);
}

// ---------------- dense: out = l2norm( lrelu((f*agg)@W1+b1) + lrelu(agg@W2+b2) ) ----------------
// 128 threads = 4 waves; each wave owns a 16-col output tile; 16 rows per tile.
__global__ __launch_bounds__(128)
void dense_kernel(const float* __restrict__ feat,
                  const float* __restrict__ agg,
                  const float* __restrict__ W1,
                  const float* __restrict__ W2,
                  const float* __restrict__ b1,
                  const float* __restrict__ b2,
                  float* __restrict__ out, int nNodes) {
    __shared__ float sW1[D * 65];
    __shared__ float sW2[D * 65];
    __shared__ float sX1[16 * 66];   // feat*agg tile (stride 66 keeps 8B align + no bank clash)
    __shared__ float sX2[16 * 66];   // agg tile
    __shared__ float sO [16 * 65];
    __shared__ float sScale[16];

    const int t    = threadIdx.x;
    const int lane = t & 31;
    const int wave = t >> 5;          // 0..3
    const int colBase = wave * 16;
    const int lm = lane & 15;         // 0..15
    const int lh = lane >> 4;         // 0 or 1

    // stage weights (64x64 each) into LDS, padded stride 65
    for (int i = t; i < D * D; i += 128) {
        int r = i >> 6, c = i & 63;
        sW1[r * 65 + c] = W1[i];
        sW2[r * 65 + c] = W2[i];
    }
    const float bb1 = b1[colBase + lm];
    const float bb2 = b2[colBase + lm];
    __syncthreads();

    const int tileBase = blockIdx.x * TILES_PER_BLOCK;
    for (int tile = 0; tile < TILES_PER_BLOCK; ++tile) {
        const int nodeBase = (tileBase + tile) * 16;
        if (nodeBase >= nNodes) break;

        // stage 16x64 A tiles (guarded rows -> 0)
        for (int i = t; i < 16 * D; i += 128) {
            int r = i >> 6, c = i & 63;
            int node = nodeBase + r;
            float f = 0.0f, a = 0.0f;
            if (node < nNodes) {
                f = feat[(size_t)node * D + c];
                a = agg [(size_t)node * D + c];
            }
            sX1[r * 66 + c] = f * a;
            sX2[r * 66 + c] = a;
        }
        __syncthreads();

        // two accumulators: c1 = (f*agg)@W1 tile, c2 = agg@W2 tile
        v8f c1 = {}; v8f c2 = {};
#pragma unroll
        for (int kk = 0; kk < 16; ++kk) {
            const int k0 = kk * 4 + lh * 2;   // A layout: lanes<16 -> K=k0,k0+1 ; lanes>=16 -> K=k0+2,k0+3
            v2f a1 = *(const v2f*)&sX1[lm * 66 + k0];
            v2f a2 = *(const v2f*)&sX2[lm * 66 + k0];
            v2f bw1, bw2;                      // B layout mirrors A: rows k0/k0+1 per lane-half
            bw1.x = sW1[(k0    ) * 65 + colBase + lm];
            bw1.y = sW1[(k0 + 1) * 65 + colBase + lm];
            bw2.x = sW2[(k0    ) * 65 + colBase + lm];
            bw2.y = sW2[(k0 + 1) * 65 + colBase + lm];
            c1 = __builtin_amdgcn_wmma_f32_16x16x4_f32(false, a1, false, bw1, (short)0, c1, false, false);
            c2 = __builtin_amdgcn_wmma_f32_16x16x4_f32(false, a2, false, bw2, (short)0, c2, false, false);
        }

        // bias + leaky relu + sum, write tile to LDS in C/D layout
#pragma unroll
        for (int r = 0; r < 8; ++r) {
            float x1 = c1[r] + bb1;
            float x2 = c2[r] + bb2;
            x1 = (x1 > 0.0f) ? x1 : ALPHA * x1;
            x2 = (x2 > 0.0f) ? x2 : ALPHA * x2;
            int row = r + lh * 8;
            sO[row * 65 + colBase + lm] = x1 + x2;
        }
        __syncthreads();

        // per-row L2 norm scale
        if (t < 16) {
            float sq = 0.0f;
            for (int c = 0; c < D; ++c) {
                float v = sO[t * 65 + c];
                sq += v * v;
            }
            sScale[t] = rsqrtf(fmaxf(sq, EPS));
        }
        __syncthreads();

        // coalesced normalized store
        for (int i = t; i < 16 * D; i += 128) {
            int r = i >> 6, c = i & 63;
            int node = nodeBase + r;
            if (node < nNodes) out[(size_t)node * D + c] = sO[r * 65 + c] * sScale[r];
        }
        __syncthreads();
    }
}

extern "C" void kernel_launch(void* const* d_in, const int* in_sizes, int n_in,
                              void* d_out, int out_size, void* d_ws, size_t ws_size,
                              hipStream_t stream) {
    const float* feat = (const float*)d_in[0];
    const int*   src  = (const int*)d_in[1];
    const int*   dst  = (const int*)d_in[2];
    const float* ew   = (const float*)d_in[3];
    const float* W1   = (const float*)d_in[4];
    const float* W2   = (const float*)d_in[5];
    const float* b1   = (const float*)d_in[6];
    const float* b2   = (const float*)d_in[7];
    float* out = (float*)d_out;

    const int N = in_sizes[0] / D;
    const int E = in_sizes[1];
    float* agg = (float*)d_ws;   // N*D floats of scratch

    const int totalAgg = N * D;
    zero_kernel<<<(totalAgg + 1023) / 1024, 256, 0, stream>>>(agg, totalAgg);

    const int edgesPerBlock = 256 / 32;
    spmm_kernel<<<(E + edgesPerBlock - 1) / edgesPerBlock, 256, 0, stream>>>(
        feat, src, dst, ew, agg, E);

    const int nodesPerBlock = 16 * TILES_PER_BLOCK;
    dense_kernel<<<(N + nodesPerBlock - 1) / nodesPerBlock, 128, 0, stream>>>(
        feat, agg, W1, W2, b1, b2, out, N);
}